// GaussianSSL_84464826843450
// MI455X (gfx1250) — compile-verified
//
#include <hip/hip_runtime.h>
#include <hip/hip_bf16.h>
#include <math.h>

typedef __attribute__((ext_vector_type(2))) float v2f;
typedef __attribute__((ext_vector_type(8))) float v8f;

#define TRI(r, c) (((r) * ((r) + 1)) / 2 + (c))

static __device__ __forceinline__ v8f wmma_f32(v2f a, v2f b, v8f c) {
  // V_WMMA_F32_16X16X4_F32: (neg_a, A, neg_b, B, c_mod, C, reuse_a, reuse_b)
  return __builtin_amdgcn_wmma_f32_16x16x4_f32(false, a, false, b, (short)0, c,
                                               false, false);
}

static __device__ __forceinline__ float fast_sigmoid(float x) {
  return 1.0f / (1.0f + __expf(-x));
}
static __device__ __forceinline__ float fast_tanh(float x) {
  float t = __expf(-2.0f * fabsf(x));
  float r = (1.0f - t) / (1.0f + t);
  return copysignf(r, x);
}

// In-place Cholesky of packed-lower 8x8 SPD matrix (36 floats).
static __device__ __forceinline__ void chol8(float* S) {
#pragma unroll
  for (int j = 0; j < 8; ++j) {
    float d = S[TRI(j, j)];
#pragma unroll
    for (int k = 0; k < j; ++k) d = fmaf(-S[TRI(j, k)], S[TRI(j, k)], d);
    float l = sqrtf(fmaxf(d, 1e-20f));
    S[TRI(j, j)] = l;
    float inv = 1.0f / l;
#pragma unroll
    for (int i = j + 1; i < 8; ++i) {
      float s = S[TRI(i, j)];
#pragma unroll
      for (int k = 0; k < j; ++k) s = fmaf(-S[TRI(i, k)], S[TRI(j, k)], s);
      S[TRI(i, j)] = s * inv;
    }
  }
}

__global__ __launch_bounds__(128) void GaussianSSL_cdna5_kernel(
    const float* __restrict__ h_in, const float* __restrict__ c_in,
    const float* __restrict__ z_prev, const float* __restrict__ x_t,
    const float* __restrict__ W_kernel, const float* __restrict__ W_rec,
    const float* __restrict__ b_lstm, const float* __restrict__ W_mu,
    const float* __restrict__ b_mu_g, const float* __restrict__ W_ls,
    const float* __restrict__ b_ls_g, const float* __restrict__ Cmat,
    const float* __restrict__ b_em, const float* __restrict__ logR,
    float* __restrict__ out, long B) {
  __shared__ float lds_hn[128 * 64];   // h_new staging  (32 KB)
  __shared__ float lds_T[128 * 16];    // [mu | log_sigma] pre-bias (8 KB)
  __shared__ float lds_Wms[64 * 16];   // [W_mu | W_log_sigma] packed (4 KB)
  __shared__ float lds_C[16 * 8];
  __shared__ float lds_rinv6[16];
  __shared__ float lds_rinv8[16];
  __shared__ float lds_G6[64];
  __shared__ float lds_G8[64];
  __shared__ float lds_be[16];
  __shared__ float lds_bmu[8];
  __shared__ float lds_bls[8];
  __shared__ float lds_logdetR;

  const int tid = threadIdx.x;
  const int lane = tid & 31;
  const int wv = tid >> 5;   // wave id == unit tile (0..3)
  const int half = lane >> 4;
  const int lc = lane & 15;
  const long blk_row0 = (long)blockIdx.x * 128;

  // ---------------- constant prep ----------------
  if (tid < 128) lds_C[tid] = Cmat[tid];
  if (tid < 16) {
    float R = __expf(logR[tid]);
    lds_rinv6[tid] = 1.0f / (R + 1e-6f);
    lds_rinv8[tid] = 1.0f / (R + 1e-8f);
    lds_be[tid] = b_em[tid];
  } else if (tid < 24) {
    lds_bmu[tid - 16] = b_mu_g[tid - 16];
  } else if (tid < 32) {
    lds_bls[tid - 24] = b_ls_g[tid - 24];
  }
  for (int i2 = tid; i2 < 64 * 16; i2 += 128) {
    int row = i2 >> 4, col = i2 & 15;
    lds_Wms[i2] = (col < 8) ? W_mu[row * 8 + col] : W_ls[row * 8 + col - 8];
  }
  __syncthreads();
  if (tid < 64) {
    int k = tid >> 3, l = tid & 7;
    float s6 = 0.0f, s8 = 0.0f;
#pragma unroll
    for (int i = 0; i < 16; ++i) {
      float ck = lds_C[i * 8 + k], cl = lds_C[i * 8 + l];
      s6 = fmaf(ck * cl, lds_rinv6[i], s6);
      s8 = fmaf(ck * cl, lds_rinv8[i], s8);
    }
    lds_G6[tid] = s6;
    lds_G8[tid] = s8;
  }
  if (tid == 0) {
    float s = 0.0f;
#pragma unroll
    for (int i = 0; i < 16; ++i) s -= __logf(lds_rinv6[i]);
    lds_logdetR = s;
  }
  __syncthreads();

  float* out_alpha = out;
  float* out_gamma = out + B;
  float* out_V = out + 9l * B;
  float* out_h = out + 73l * B;
  float* out_c = out + 137l * B;

  // gate biases for this wave's 16 units (uniform over row tiles)
  const int unit = wv * 16 + lc;
  const float bl0 = b_lstm[0 * 64 + unit];
  const float bl1 = b_lstm[1 * 64 + unit];
  const float bl2 = b_lstm[2 * 64 + unit];
  const float bl3 = b_lstm[3 * 64 + unit];

  // ---------------- Phase A: LSTM gate GEMM via WMMA f32 16x16x4 ----------
  for (int rt = 0; rt < 8; ++rt) {
    const long row_base = blk_row0 + rt * 16;
    const long arow = row_base + lc;
    // A fragments: A = [z_prev (K=0..7) | h (K=8..71)], 18 K-steps of 4
    v2f af[18];
#pragma unroll
    for (int kk = 0; kk < 2; ++kk) {
      int k = kk * 4 + half * 2;
      af[kk] = *(const v2f*)(z_prev + arow * 8 + k);
    }
#pragma unroll
    for (int kk = 2; kk < 18; ++kk) {
      int k = (kk - 2) * 4 + half * 2;
      af[kk] = *(const v2f*)(h_in + arow * 64 + k);
    }
    v8f acc0 = {}, acc1 = {}, acc2 = {}, acc3 = {};
#pragma unroll
    for (int kk = 0; kk < 18; ++kk) {
      const float* wp;
      if (kk < 2)
        wp = W_kernel + (kk * 4 + 2 * half) * 256 + unit;
      else
        wp = W_rec + ((kk - 2) * 4 + 2 * half) * 256 + unit;
      v2f b0, b1, b2, b3;
      b0.x = wp[0];       b0.y = wp[256];
      b1.x = wp[64];      b1.y = wp[64 + 256];
      b2.x = wp[128];     b2.y = wp[128 + 256];
      b3.x = wp[192];     b3.y = wp[192 + 256];
      acc0 = wmma_f32(af[kk], b0, acc0);
      acc1 = wmma_f32(af[kk], b1, acc1);
      acc2 = wmma_f32(af[kk], b2, acc2);
      acc3 = wmma_f32(af[kk], b3, acc3);
    }
    // activations + state update, all four gates in-register
#pragma unroll
    for (int v = 0; v < 8; ++v) {
      long row = row_base + v + 8 * half;
      float iv = fast_sigmoid(acc0[v] + bl0);
      float fv = fast_sigmoid(acc1[v] + bl1);
      float gv = fast_tanh(acc2[v] + bl2);
      float ov = fast_sigmoid(acc3[v] + bl3);
      float cold = c_in[row * 64 + unit];
      float cn = fmaf(fv, cold, iv * gv);
      float hn = ov * fast_tanh(cn);
      out_c[row * 64 + unit] = cn;
      out_h[row * 64 + unit] = hn;
      lds_hn[(rt * 16 + v + 8 * half) * 64 + unit] = hn;
    }
  }
  __syncthreads();

  // ------------- Phase B: T = h_new @ [W_mu|W_log_sigma]  (K=64) ----------
#pragma unroll
  for (int t = 0; t < 2; ++t) {
    const int rt2 = wv + t * 4;  // 8 row tiles over 4 waves x 2
    v8f acc = {};
#pragma unroll
    for (int kk = 0; kk < 16; ++kk) {
      int k = kk * 4 + half * 2;
      v2f a = *(const v2f*)&lds_hn[(rt2 * 16 + lc) * 64 + k];
      v2f b;
      b.x = lds_Wms[(kk * 4 + 2 * half) * 16 + lc];
      b.y = lds_Wms[(kk * 4 + 2 * half + 1) * 16 + lc];
      acc = wmma_f32(a, b, acc);
    }
#pragma unroll
    for (int v = 0; v < 8; ++v)
      lds_T[(rt2 * 16 + v + 8 * half) * 16 + lc] = acc[v];
  }
  __syncthreads();

  // ------------- Phase C: per-row 8x8 Woodbury/Cholesky work --------------
  {
    const long b = blk_row0 + tid;
    float mu[8], sg[8];
#pragma unroll
    for (int k = 0; k < 8; ++k) {
      mu[k] = lds_T[tid * 16 + k] + lds_bmu[k];
      float ls = lds_T[tid * 16 + 8 + k] + lds_bls[k];
      ls = fminf(3.0f, fmaxf(-5.0f, ls));
      sg[k] = __expf(ls);
    }
    float vv[8], infoacc[8];
#pragma unroll
    for (int k = 0; k < 8; ++k) { vv[k] = 0.0f; infoacc[k] = 0.0f; }
    float dTr = 0.0f;
#pragma unroll
    for (int i = 0; i < 16; ++i) {
      float pm = lds_be[i];
#pragma unroll
      for (int k = 0; k < 8; ++k) pm = fmaf(lds_C[i * 8 + k], mu[k], pm);
      float xi = x_t[b * 16 + i];
      float d = xi - pm;
      float w = d * lds_rinv6[i];
      dTr = fmaf(d, w, dTr);
      float xe = (xi - lds_be[i]) * lds_rinv8[i];
#pragma unroll
      for (int k = 0; k < 8; ++k) {
        vv[k] = fmaf(lds_C[i * 8 + k], w, vv[k]);
        infoacc[k] = fmaf(lds_C[i * 8 + k], xe, infoacc[k]);
      }
    }
    // ---- alpha: S = I + (sg sg^T) .* G6 ; mahal = dTr - ||L^-1 u||^2
    float S[36];
#pragma unroll
    for (int k = 0; k < 8; ++k)
#pragma unroll
      for (int l = 0; l <= k; ++l)
        S[TRI(k, l)] = sg[k] * sg[l] * lds_G6[k * 8 + l] + (k == l ? 1.0f : 0.0f);
    chol8(S);
    float logdetS = 0.0f;
#pragma unroll
    for (int j = 0; j < 8; ++j) logdetS += 2.0f * __logf(S[TRI(j, j)]);
    float y[8];
#pragma unroll
    for (int i = 0; i < 8; ++i) {
      float s = sg[i] * vv[i];  // u_i
#pragma unroll
      for (int k = 0; k < i; ++k) s = fmaf(-S[TRI(i, k)], y[k], s);
      y[i] = s / S[TRI(i, i)];
    }
    float mahal = dTr;
#pragma unroll
    for (int i = 0; i < 8; ++i) mahal = fmaf(-y[i], y[i], mahal);
    const float LOG2PI = 1.8378770664093453f;
    out_alpha[b] = -0.5f * (mahal + logdetS + lds_logdetR + 16.0f * LOG2PI);

    // ---- gamma: V = (G8 + diag(1/(sg^2+1e-8) + 1e-6))^-1 ; gamma = V info
    float Sinv[8], info[8];
#pragma unroll
    for (int k = 0; k < 8; ++k) {
      Sinv[k] = 1.0f / (sg[k] * sg[k] + 1e-8f);
      info[k] = fmaf(Sinv[k], mu[k], infoacc[k]);
    }
    float A2[36];
#pragma unroll
    for (int k = 0; k < 8; ++k)
#pragma unroll
      for (int l = 0; l <= k; ++l)
        A2[TRI(k, l)] = lds_G8[k * 8 + l] + (k == l ? (Sinv[k] + 1e-6f) : 0.0f);
    chol8(A2);
    // J = L^-1 (packed lower)
    float J[36];
#pragma unroll
    for (int j = 0; j < 8; ++j) {
      J[TRI(j, j)] = 1.0f / A2[TRI(j, j)];
#pragma unroll
      for (int i = j + 1; i < 8; ++i) {
        float s = 0.0f;
#pragma unroll
        for (int k = j; k < i; ++k) s = fmaf(-A2[TRI(i, k)], J[TRI(k, j)], s);
        J[TRI(i, j)] = s / A2[TRI(i, i)];
      }
    }
    // V[k][l] = sum_{m>=max(k,l)} J[m][k] * J[m][l]; gamma += V * info
    float gamma[8];
#pragma unroll
    for (int k = 0; k < 8; ++k) gamma[k] = 0.0f;
#pragma unroll
    for (int k = 0; k < 8; ++k) {
#pragma unroll
      for (int l = 0; l <= k; ++l) {
        float s = 0.0f;
#pragma unroll
        for (int m = 0; m < 8; ++m)
          if (m >= k) s = fmaf(J[TRI(m, k)], J[TRI(m, l)], s);
        out_V[b * 64 + k * 8 + l] = s;
        if (k != l) out_V[b * 64 + l * 8 + k] = s;
        gamma[k] = fmaf(s, info[l], gamma[k]);
        if (k != l) gamma[l] = fmaf(s, info[k], gamma[l]);
      }
    }
#pragma unroll
    for (int k = 0; k < 8; ++k) out_gamma[b * 8 + k] = gamma[k];
  }
}

extern "C" void kernel_launch(void* const* d_in, const int* in_sizes, int n_in,
                              void* d_out, int out_size, void* d_ws,
                              size_t ws_size, hipStream_t stream) {
  (void)n_in; (void)out_size; (void)d_ws; (void)ws_size;
  const float* h = (const float*)d_in[0];
  const float* c = (const float*)d_in[1];
  const float* z_prev = (const float*)d_in[2];
  const float* x_t = (const float*)d_in[3];
  const float* W_kernel = (const float*)d_in[4];
  const float* W_rec = (const float*)d_in[5];
  const float* b_lstm = (const float*)d_in[6];
  const float* W_mu = (const float*)d_in[7];
  const float* b_mu = (const float*)d_in[8];
  const float* W_ls = (const float*)d_in[9];
  const float* b_ls = (const float*)d_in[10];
  const float* C = (const float*)d_in[11];
  const float* b_em = (const float*)d_in[12];
  const float* logR = (const float*)d_in[13];
  long B = in_sizes[0] / 64;  // h is [B, 64]
  dim3 grid((unsigned)(B / 128)), block(128);
  hipLaunchKernelGGL(GaussianSSL_cdna5_kernel, grid, block, 0, stream, h, c,
                     z_prev, x_t, W_kernel, W_rec, b_lstm, W_mu, b_mu, W_ls,
                     b_ls, C, b_em, logR, (float*)d_out, B);
}